// MultiResEncoding_23313082483304
// MI455X (gfx1250) — compile-verified
//
#include <hip/hip_runtime.h>
#include <hip/hip_bf16.h>

// Instant-NGP style 2D multires hash-grid encoding for MI455X (gfx1250).
//
// Mapping: one thread per (point, level). g = global thread id,
//   p = g >> 4, l = g & 15  ->  output float2 index is exactly g
// so output stores are perfectly linear per wave (256B per wave32 store).
//
// Memory policy (the actual optimization on this chip):
//   - output (256 MB, write-once)  : non-temporal stores (TH_STORE_NT)
//   - coords (16 MB, read-once)    : non-temporal loads
//   - hash tables (60 MB)          : default RT policy -> stay resident in 192MB L2
//   - level-0 dense grid (2 KB)    : staged into LDS via gfx1250 async global->LDS

typedef float fv2 __attribute__((ext_vector_type(2)));
typedef int   iv4 __attribute__((ext_vector_type(4)));

namespace {

constexpr int      kLevels    = 16;
constexpr unsigned kTableSize = 1u << 19;          // power of two -> AND instead of %
constexpr unsigned kPrimeY    = 2654435761u;       // PRIMES = (1, 2654435761)

// floor(16 * 2^(7l/15)) for l = 0..15 (matches reference _level_res())
__device__ __constant__ int kRes[kLevels] = {
    16, 22, 30, 42, 58, 80, 111, 153, 212, 294, 406, 561, 776, 1072, 1482, 2048};

__device__ __forceinline__ fv2 hash_gather(const float* __restrict__ t,
                                           unsigned ix, unsigned iy) {
  unsigned h   = ix ^ (iy * kPrimeY);              // prime for x is 1
  unsigned idx = h & (kTableSize - 1u);
  return *reinterpret_cast<const fv2*>(t + (size_t)idx * 2u);
}

} // namespace

extern "C" __global__ __launch_bounds__(256)
void mre_hashgrid_kernel(const float* __restrict__ cx,
                         const float* __restrict__ cy,
                         const float* __restrict__ dense,
                         const float* __restrict__ hashes,
                         float* __restrict__ out,
                         unsigned npoints) {
  __shared__ float lds_dense[16 * 16 * 2];         // 2 KB level-0 grid

  // ---- Stage dense level-0 table into LDS --------------------------------
#if __has_builtin(__builtin_amdgcn_global_load_async_to_lds_b128)
  if (threadIdx.x < 128u) {                        // 128 lanes x 16B = 2 KB
    const float* gp = dense + threadIdx.x * 4u;
    float*       lp = &lds_dense[threadIdx.x * 4u];
    typedef __attribute__((address_space(1))) iv4* g_iv4_p;
    typedef __attribute__((address_space(3))) iv4* l_iv4_p;
    // generic->AS1 is numerically identity; generic LDS addr low 32 bits are
    // the raw LDS offset (ISA 10.2), so integer round-trips are safe.
    __builtin_amdgcn_global_load_async_to_lds_b128(
        (g_iv4_p)(uintptr_t)gp, (l_iv4_p)(uintptr_t)lp, 0, 0);
  }
  asm volatile("s_wait_asynccnt 0" ::: "memory");
  __syncthreads();
#else
  for (unsigned i = threadIdx.x; i < 512u; i += blockDim.x)
    lds_dense[i] = dense[i];
  __syncthreads();
#endif

  // ---- Per-(point, level) work -------------------------------------------
  unsigned g = blockIdx.x * 256u + threadIdx.x;
  unsigned p = g >> 4;
  if (p >= npoints) return;
  unsigned l = g & 15u;

  // coords are read-once (16 consecutive lanes share one point): NT loads
  float x = __builtin_nontemporal_load(cx + p);
  float y = __builtin_nontemporal_load(cy + p);

  int   r   = kRes[l];
  int   rm1 = r - 1;
  float fr  = (float)rm1;
  float fx  = x * fr;
  float fy  = y * fr;
  float x0f = floorf(fx);
  float y0f = floorf(fy);
  float wx  = fx - x0f;                            // weights from unclamped frac
  float wy  = fy - y0f;

  int ix0 = (int)x0f; ix0 = ix0 < 0 ? 0 : (ix0 > rm1 ? rm1 : ix0);
  int iy0 = (int)y0f; iy0 = iy0 < 0 ? 0 : (iy0 > rm1 ? rm1 : iy0);
  int ix1 = (ix0 + 1 > rm1) ? rm1 : ix0 + 1;
  int iy1 = (iy0 + 1 > rm1) ? rm1 : iy0 + 1;

  fv2 f00, f01, f10, f11;
  if (l == 0u) {
    // dense 16x16 grid from LDS (ds_load_b64)
    const fv2* dt = reinterpret_cast<const fv2*>(lds_dense);
    f00 = dt[ix0 * 16 + iy0];
    f01 = dt[ix0 * 16 + iy1];
    f10 = dt[ix1 * 16 + iy0];
    f11 = dt[ix1 * 16 + iy1];
  } else {
    // hash levels: tables stay RT-cached in L2 (60 MB << 192 MB)
    const float* t = hashes + (size_t)(l - 1u) * (size_t)(kTableSize * 2u);
    f00 = hash_gather(t, (unsigned)ix0, (unsigned)iy0);
    f01 = hash_gather(t, (unsigned)ix0, (unsigned)iy1);
    f10 = hash_gather(t, (unsigned)ix1, (unsigned)iy0);
    f11 = hash_gather(t, (unsigned)ix1, (unsigned)iy1);
  }

  float w00 = (1.0f - wx) * (1.0f - wy);
  float w01 = (1.0f - wx) * wy;
  float w10 = wx * (1.0f - wy);
  float w11 = wx * wy;

  fv2 res;
  res.x = f00.x * w00 + f01.x * w01 + f10.x * w10 + f11.x * w11;
  res.y = f00.y * w00 + f01.y * w01 + f10.y * w10 + f11.y * w11;

  // write-once 256 MB output: NT store, perfectly linear (fv2 offset == g)
  __builtin_nontemporal_store(res, reinterpret_cast<fv2*>(out) + g);
}

extern "C" void kernel_launch(void* const* d_in, const int* in_sizes, int n_in,
                              void* d_out, int out_size, void* d_ws, size_t ws_size,
                              hipStream_t stream) {
  const float* cx     = (const float*)d_in[0];   // [N]
  const float* cy     = (const float*)d_in[1];   // [N]
  const float* dense  = (const float*)d_in[2];   // [16,16,2]
  const float* hashes = (const float*)d_in[3];   // [15, 2^19, 2]
  float*       out    = (float*)d_out;           // [N, 32] floats

  unsigned n = (unsigned)in_sizes[0];
  unsigned long long total  = (unsigned long long)n * 16ull;
  unsigned           blocks = (unsigned)((total + 255ull) / 256ull);

  mre_hashgrid_kernel<<<dim3(blocks), dim3(256), 0, stream>>>(
      cx, cy, dense, hashes, out, n);
  (void)d_ws; (void)ws_size; (void)out_size; (void)n_in;
}